// SSIMcal_71365176590844
// MI455X (gfx1250) — compile-verified
//
#include <hip/hip_runtime.h>

// SSIM (7x7 uniform window) for B=32, 512x512 f32 images on gfx1250.
// Bandwidth-bound (~67MB reads -> ~2.9us at 23.3TB/s). Box filters run as
// banded-ones f32 WMMAs (v_wmma_f32_16x16x4_f32) on the matrix pipe (co-
// executes with VALU); halo tiles stream global->LDS with
// global_load_async_to_lds_b32 (ASYNCcnt), zero-padded at image borders.

typedef __attribute__((ext_vector_type(2))) float v2f;
typedef __attribute__((ext_vector_type(8))) float v8f;

#define IMG_H 512
#define IMG_W 512
#define HALO 3            // (7-1)/2
#define TILE_W 64         // per workgroup
#define TILE_H 32
#define TILES_X (IMG_W / TILE_W)                 // 8
#define TILES_Y (IMG_H / TILE_H)                 // 16
#define TILES_PER_IMG (TILES_X * TILES_Y)        // 128
#define LDS_W 72          // 70 used (64+6), padded
#define LDS_H 38          // 32+6 rows loaded from global
#define LDS_H2 40         // +2 zeroed rows so K-pad reads need no gating
#define VBUF_W 34         // 32 used (two 16-col chunks) + 2 zeroed pad cols

// plane p value from the two shared tiles
__device__ __forceinline__ float plane_val(const float sI[LDS_H2][LDS_W],
                                           const float sR[LDS_H2][LDS_W],
                                           int p, int y, int x) {
  float a = sI[y][x];
  float b = sR[y][x];
  switch (p) {
    case 0:  return a;        // x
    case 1:  return b;        // y
    case 2:  return a * a;    // xx
    case 3:  return b * b;    // yy
    default: return a * b;    // xy
  }
}

__global__ __launch_bounds__(256)
void ssim_tile_kernel(const float* __restrict__ img,
                      const float* __restrict__ ref,
                      const float* __restrict__ drange,
                      const float* __restrict__ w,
                      float* __restrict__ partials) {
  __shared__ float sImg[LDS_H2][LDS_W];
  __shared__ float sRef[LDS_H2][LDS_W];
  __shared__ float vbuf[8][16][VBUF_W];  // per-wave vertical-pass scratch
  __shared__ float red[256];

  const int tid  = threadIdx.x;
  const int b    = blockIdx.z;
  const int row0 = blockIdx.y * TILE_H;
  const int col0 = blockIdx.x * TILE_W;

  const float* imgB = img + (size_t)b * (IMG_H * IMG_W);
  const float* refB = ref + (size_t)b * (IMG_H * IMG_W);

  // ---- Stage 0: halo load, global -> LDS async (zero pad = SAME conv) ----
  for (int i = tid; i < LDS_H * 70; i += 256) {
    int y = i / 70, x = i % 70;
    int gy = row0 - HALO + y;
    int gx = col0 - HALO + x;
    if (gy >= 0 && gy < IMG_H && gx >= 0 && gx < IMG_W) {
      int goff = (gy * IMG_W + gx) * 4;                     // byte offset
      unsigned ldsI = (unsigned)(size_t)&sImg[y][x];        // LDS byte addr
      unsigned ldsR = (unsigned)(size_t)&sRef[y][x];
      asm volatile("global_load_async_to_lds_b32 %0, %1, %2"
                   :: "v"(ldsI), "v"(goff), "s"(imgB) : "memory");
      asm volatile("global_load_async_to_lds_b32 %0, %1, %2"
                   :: "v"(ldsR), "v"(goff), "s"(refB) : "memory");
    } else {
      sImg[y][x] = 0.0f;
      sRef[y][x] = 0.0f;
    }
  }
  // K-pad rows: zeroed so vertical-pass reads need no gating
  for (int i = tid; i < (LDS_H2 - LDS_H) * 70; i += 256) {
    int y = LDS_H + i / 70, x = i % 70;
    sImg[y][x] = 0.0f;
    sRef[y][x] = 0.0f;
  }
  asm volatile("s_wait_asynccnt 0x0" ::: "memory");
  __syncthreads();

  // ---- Per-wave 16x16 tile setup (wave32) ----
  const int wave = tid >> 5;
  const int lane = tid & 31;
  const int m16  = lane & 15;        // row (A) / col (B,C/D) index
  const int hh   = lane >> 4;        // lane-half selects K pair
  const int wty  = (wave >> 2) * 16; // wave tile origin inside block tile
  const int wtx  = (wave & 3) * 16;

  // zero vbuf pad cols 32,33 once (K=22,23 pad reads hit these)
  vbuf[wave][m16][32 + hh] = 0.0f;

  // Band fragments, shared by vertical-A and horizontal-B operands:
  // band(k) = 1 if m16 <= k <= m16+6. Built once, reused for all 90 WMMAs.
  v2f band[6];
#pragma unroll
  for (int j = 0; j < 6; ++j) {
    const int k0 = 4 * j + 2 * hh;
    band[j].x = (k0     >= m16 && k0     <= m16 + 6) ? 1.0f : 0.0f;
    band[j].y = (k0 + 1 >= m16 && k0 + 1 <= m16 + 6) ? 1.0f : 0.0f;
  }

  v8f facc[5];                        // final filtered window sums per plane

#pragma unroll
  for (int p = 0; p < 5; ++p) {
    // ---- Vertical pass: Band(16x24 ones) x Data(24x16), two col chunks ----
#pragma unroll
    for (int cc = 0; cc < 2; ++cc) {
      v8f acc = {};
      const int xo = wtx + 6 * cc + m16;   // LDS x of this B column
#pragma unroll
      for (int j = 0; j < 6; ++j) {
        const int k0 = 4 * j + 2 * hh;
        // B = plane data rows k0,k0+1 (rows 38,39 are zeroed pad)
        v2f bb;
        bb.x = plane_val(sImg, sRef, p, wty + k0,     xo);
        bb.y = plane_val(sImg, sRef, p, wty + k0 + 1, xo);
        acc = __builtin_amdgcn_wmma_f32_16x16x4_f32(
            false, band[j], false, bb, (short)0, acc, false, false);
      }
      // store chunk: chunk0 -> cols 0..15 (j=n), chunk1 -> cols 16..31 (j=6+n)
#pragma unroll
      for (int v = 0; v < 8; ++v) {
        vbuf[wave][v + 8 * hh][cc * 16 + m16] = acc[v];
      }
    }
    // same-wave LDS store->load ordering (cross-lane round trip)
    asm volatile("s_wait_dscnt 0x0" ::: "memory");

    // ---- Horizontal pass: Vsum(16x24) x Band(24x16) ----
    {
      v8f hacc = {};
#pragma unroll
      for (int j = 0; j < 6; ++j) {
        const int k0 = 4 * j + 2 * hh;
        // A = vsum(m16, k): k<16 at col k, else col k+10 (22,23 -> zero pad)
        v2f a;
        a.x = vbuf[wave][m16][(k0     < 16) ? k0     : k0 + 10];
        a.y = vbuf[wave][m16][(k0 + 1 < 16) ? k0 + 1 : k0 + 11];
        hacc = __builtin_amdgcn_wmma_f32_16x16x4_f32(
            false, a, false, band[j], (short)0, hacc, false, false);
      }
      facc[p] = hacc;
    }
    // keep vbuf reuse ordered across planes
    asm volatile("s_wait_dscnt 0x0" ::: "memory");
  }

  // ---- Pointwise SSIM + per-thread partial sum ----
  const float w0 = w[0];                    // 1/49 (uniform window weight)
  const float dr = drange[b];
  const float C1 = (0.01f * dr) * (0.01f * dr);
  const float C2 = (0.03f * dr) * (0.03f * dr);
  const float CN = 49.0f / 48.0f;           // NP_WIN / (NP_WIN - 1)

  float ssum = 0.0f;
#pragma unroll
  for (int e = 0; e < 8; ++e) {
    float ux  = w0 * facc[0][e];
    float uy  = w0 * facc[1][e];
    float uxx = w0 * facc[2][e];
    float uyy = w0 * facc[3][e];
    float uxy = w0 * facc[4][e];
    float vx  = CN * (uxx - ux * ux);
    float vy  = CN * (uyy - uy * uy);
    float vxy = CN * (uxy - ux * uy);
    float A1 = 2.0f * ux * uy + C1;
    float A2 = 2.0f * vxy + C2;
    float B1 = ux * ux + uy * uy + C1;
    float B2 = vx + vy + C2;
    ssum += (A1 * A2) / (B1 * B2);
  }

  // ---- Deterministic block tree reduction ----
  red[tid] = ssum;
  __syncthreads();
#pragma unroll
  for (int s = 128; s > 0; s >>= 1) {
    if (tid < s) red[tid] += red[tid + s];
    __syncthreads();
  }
  if (tid == 0) {
    partials[(size_t)b * TILES_PER_IMG + blockIdx.y * TILES_X + blockIdx.x] = red[0];
  }
}

__global__ __launch_bounds__(128)
void ssim_reduce_kernel(const float* __restrict__ partials,
                        float* __restrict__ out) {
  __shared__ float red[128];
  const int b = blockIdx.x;
  red[threadIdx.x] = partials[(size_t)b * TILES_PER_IMG + threadIdx.x];
  __syncthreads();
#pragma unroll
  for (int s = 64; s > 0; s >>= 1) {
    if (threadIdx.x < s) red[threadIdx.x] += red[threadIdx.x + s];
    __syncthreads();
  }
  if (threadIdx.x == 0) {
    out[b] = red[0] * (1.0f / (float)(IMG_H * IMG_W));
  }
}

extern "C" void kernel_launch(void* const* d_in, const int* in_sizes, int n_in,
                              void* d_out, int out_size, void* d_ws, size_t ws_size,
                              hipStream_t stream) {
  const float* img    = (const float*)d_in[0];
  const float* ref    = (const float*)d_in[1];
  const float* drange = (const float*)d_in[2];
  const float* w      = (const float*)d_in[3];
  float* out      = (float*)d_out;
  float* partials = (float*)d_ws;   // 32 * 128 floats = 16 KB

  dim3 grid(TILES_X, TILES_Y, 32);
  ssim_tile_kernel<<<grid, 256, 0, stream>>>(img, ref, drange, w, partials);
  ssim_reduce_kernel<<<32, 128, 0, stream>>>(partials, out);
}